// Net_72009421685167
// MI455X (gfx1250) — compile-verified
//
#include <hip/hip_runtime.h>

typedef __attribute__((ext_vector_type(2))) float v2f;
typedef __attribute__((ext_vector_type(8))) float v8f;

#define N_NODES 50000
#define N_EDGES 400000
#define IN_F    768
#define HID_F   16
#define OUT_F   21

// ---------------------------------------------------------------------------
// Layer-1 GEMM (reordered GCN):  Z[50000,16] = X[50000,768] @ W1[768,16]
// One wave32 per 16x16 output tile using V_WMMA_F32_16X16X4_F32 (full fp32).
// 50000 = 3125 * 16, so every tile is full -> EXEC all-ones for WMMA.
// W1 (48 KB) is staged in LDS once per block and reused by all 8 waves.
// ---------------------------------------------------------------------------
__global__ __launch_bounds__(256) void gcn_gemm1_wmma(
    const float* __restrict__ X, const float* __restrict__ W1,
    float* __restrict__ Z)
{
  __shared__ float w1s[IN_F * HID_F];  // 48 KB
  {
    const float4* w4 = (const float4*)W1;
    float4*       s4 = (float4*)w1s;
    const int n4 = (IN_F * HID_F) / 4;  // 3072
    for (int i = threadIdx.x; i < n4; i += 256) s4[i] = w4[i];
  }
  __syncthreads();

  const int lane = threadIdx.x & 31;
  const int wave = threadIdx.x >> 5;
  const int tile = blockIdx.x * 8 + wave;
  if (tile >= (N_NODES / 16)) return;          // wave-uniform branch
  const int m0   = tile * 16;

  const int half = lane >> 4;                  // 0: K pair {0,1}, 1: K pair {2,3}
  const int l16  = lane & 15;
  const int koff = half << 1;

  // A-matrix 16x4 fp32 layout: lane holds row (m0+l16), K = koff, koff+1
  const float* xrow = X + (size_t)(m0 + l16) * IN_F + koff;

  v8f acc0 = {};
  v8f acc1 = {};
  for (int k = 0; k < IN_F; k += 8) {
    v2f a0 = *(const v2f*)(xrow + k);          // 8B aligned (koff even, k%4==0)
    v2f a1 = *(const v2f*)(xrow + k + 4);
    v2f b0, b1;                                // B 4x16: VGPR0={K0|K2}, VGPR1={K1|K3}
    b0.x = w1s[(k + koff)     * HID_F + l16];
    b0.y = w1s[(k + koff + 1) * HID_F + l16];
    b1.x = w1s[(k + 4 + koff)     * HID_F + l16];
    b1.y = w1s[(k + 4 + koff + 1) * HID_F + l16];
    acc0 = __builtin_amdgcn_wmma_f32_16x16x4_f32(false, a0, false, b0,
                                                 (short)0, acc0, false, false);
    acc1 = __builtin_amdgcn_wmma_f32_16x16x4_f32(false, a1, false, b1,
                                                 (short)0, acc1, false, false);
  }

  // D layout: VGPR r -> row m0 + half*8 + r, col l16
  float* zp = Z + (size_t)(m0 + half * 8) * HID_F + l16;
#pragma unroll
  for (int r = 0; r < 8; ++r) zp[r * HID_F] = acc0[r] + acc1[r];
}

// ---------------------------------------------------------------------------
// Zero a float buffer
// ---------------------------------------------------------------------------
__global__ __launch_bounds__(256) void gcn_zero(float* __restrict__ p, int n)
{
  int i = blockIdx.x * 256 + threadIdx.x;
  if (i < n) p[i] = 0.0f;
}

// ---------------------------------------------------------------------------
// Edge scatter-add of 16-dim features: acc[dst[e]][f] += feat[src[e]][f]
// One lane per (edge, feature); 64B contiguous per edge -> coalesced gather,
// all buffers (3.2 MB) are L2-resident so atomics run at L2 rate.
// ---------------------------------------------------------------------------
__global__ __launch_bounds__(256) void gcn_scatter16(
    const float* __restrict__ feat, const int* __restrict__ src,
    const int* __restrict__ dst, float* __restrict__ acc)
{
  unsigned tid = blockIdx.x * 256u + threadIdx.x;
  unsigned e = tid >> 4;
  unsigned f = tid & 15;
  if (e < N_EDGES) {
    int s = src[e];
    int d = dst[e];
    atomicAdd(acc + (size_t)d * HID_F + f, feat[(size_t)s * HID_F + f]);
  }
}

// ---------------------------------------------------------------------------
// H = relu(A1 + b1)  elementwise over [50000,16]
// ---------------------------------------------------------------------------
__global__ __launch_bounds__(256) void gcn_bias_relu(
    const float* __restrict__ a, const float* __restrict__ b1,
    float* __restrict__ h)
{
  int i = blockIdx.x * 256 + threadIdx.x;
  if (i < N_NODES * HID_F) {
    float v = a[i] + b1[i & (HID_F - 1)];
    h[i] = v > 0.0f ? v : 0.0f;
  }
}

// ---------------------------------------------------------------------------
// Layer-2 GEMM: out[50000,21] = A2[50000,16] @ W2[16,21] + b2
// WMMA fp32 with N padded to two 16-wide tiles; invalid columns use B=0 and
// predicated stores (WMMA executes with full EXEC).
// ---------------------------------------------------------------------------
__global__ __launch_bounds__(256) void gcn_gemm2_wmma(
    const float* __restrict__ A, const float* __restrict__ W2,
    const float* __restrict__ b2, float* __restrict__ out)
{
  const int lane = threadIdx.x & 31;
  const int wave = threadIdx.x >> 5;
  const int t    = blockIdx.x * 8 + wave;
  const int mtiles = N_NODES / 16;             // 3125
  if (t >= mtiles * 2) return;                 // wave-uniform branch
  const int mt = t >> 1;
  const int nt = t & 1;
  const int m0 = mt * 16;

  const int half = lane >> 4;
  const int l16  = lane & 15;
  const int koff = half << 1;
  const int n    = nt * 16 + l16;
  const bool nvalid = (n < OUT_F);

  const float* arow = A + (size_t)(m0 + l16) * HID_F + koff;

  v8f acc = {};
#pragma unroll
  for (int k = 0; k < HID_F; k += 4) {
    v2f a = *(const v2f*)(arow + k);
    v2f b;
    b.x = nvalid ? W2[(k + koff)     * OUT_F + n] : 0.0f;
    b.y = nvalid ? W2[(k + koff + 1) * OUT_F + n] : 0.0f;
    acc = __builtin_amdgcn_wmma_f32_16x16x4_f32(false, a, false, b,
                                                (short)0, acc, false, false);
  }

  if (nvalid) {
    float bias = b2[n];
    float* op = out + (size_t)(m0 + half * 8) * OUT_F + n;
#pragma unroll
    for (int r = 0; r < 8; ++r) op[r * OUT_F] = acc[r] + bias;
  }
}

// ---------------------------------------------------------------------------
extern "C" void kernel_launch(void* const* d_in, const int* in_sizes, int n_in,
                              void* d_out, int out_size, void* d_ws, size_t ws_size,
                              hipStream_t stream)
{
  const float* feature = (const float*)d_in[0];
  const float* W1      = (const float*)d_in[1];
  const float* b1      = (const float*)d_in[2];
  const float* W2      = (const float*)d_in[3];
  const float* b2      = (const float*)d_in[4];
  const int*   src     = (const int*)d_in[5];
  const int*   dst     = (const int*)d_in[6];
  float*       out     = (float*)d_out;

  float* buf0 = (float*)d_ws;                          // Z1, later H   (3.2 MB)
  float* buf1 = buf0 + (size_t)N_NODES * HID_F;        // A1, later A2  (3.2 MB)

  const int NH = N_NODES * HID_F;                      // 800000
  const int zero_blocks    = (NH + 255) / 256;         // 3125
  const int scatter_blocks = (N_EDGES * HID_F) / 256;  // 25000
  const int gemm1_blocks   = ((N_NODES / 16) + 7) / 8; // 391
  const int gemm2_blocks   = ((N_NODES / 16) * 2 + 7) / 8; // 782

  // Z1 = X @ W1  (agg and linear commute: agg(X)@W1 == agg(X@W1))
  gcn_gemm1_wmma<<<gemm1_blocks, 256, 0, stream>>>(feature, W1, buf0);

  // A1 = scatter_add(Z1[src] -> dst)
  gcn_zero<<<zero_blocks, 256, 0, stream>>>(buf1, NH);
  gcn_scatter16<<<scatter_blocks, 256, 0, stream>>>(buf0, src, dst, buf1);

  // H = relu(A1 + b1)   (overwrites Z1, which is dead)
  gcn_bias_relu<<<zero_blocks, 256, 0, stream>>>(buf1, b1, buf0);

  // A2 = scatter_add(H[src] -> dst)   (overwrites A1, which is dead)
  gcn_zero<<<zero_blocks, 256, 0, stream>>>(buf1, NH);
  gcn_scatter16<<<scatter_blocks, 256, 0, stream>>>(buf0, src, dst, buf1);

  // out = A2 @ W2 + b2
  gcn_gemm2_wmma<<<gemm2_blocks, 256, 0, stream>>>(buf1, W2, b2, out);
}